// DepthAwareGAT_86002425135774
// MI455X (gfx1250) — compile-verified
//
#include <hip/hip_runtime.h>
#include <hip/hip_bf16.h>

typedef __attribute__((ext_vector_type(2))) float v2f;
typedef __attribute__((ext_vector_type(8))) float v8f;

#define SLOPE 0.2f

__device__ __forceinline__ unsigned fkey(float f) {
  unsigned u = __float_as_uint(f);
  return (u & 0x80000000u) ? ~u : (u | 0x80000000u);
}
__device__ __forceinline__ float unfkey(unsigned k) {
  unsigned u = (k & 0x80000000u) ? (k & 0x7FFFFFFFu) : ~k;
  return __uint_as_float(u);
}

// -------------------------------------------------------------------------
// C[M,N] = A[M,K] * B[K,N], row-major fp32, via V_WMMA_F32_16X16X4_F32.
// One wave computes a 16 x (16*NB) tile: the A fragment is loaded once per
// K-step (one global_load_b64 per lane) and reused across NB WMMAs.
// Requires M%16==0, N%(16*NB)==0, K%4==0.
// A frag (16x4): lane m=lane&15, khi=lane>>4; {A[m][k0+2khi], A[m][k0+2khi+1]}
// B frag (4x16): lane n=lane&15, khi=lane>>4; {B[k0+2khi][n], B[k0+2khi+1][n]}
// C/D (16x16):   VGPR r -> row (r + 8*(lane>>4)), col lane&15
// -------------------------------------------------------------------------
template <int NB>
__global__ void gemm_f32_wmma(const float* __restrict__ A,
                              const float* __restrict__ B,
                              float* __restrict__ C,
                              int M, int K, int N) {
  const int lane = threadIdx.x & 31;
  const int wave = threadIdx.x >> 5;
  const int ngroups = N / (16 * NB);
  const long tiles = (long)(M >> 4) * ngroups;
  const long t = (long)blockIdx.x * (blockDim.x >> 5) + wave;
  if (t >= tiles) return;                    // wave-uniform: EXEC stays all-1s
  const int row0 = (int)(t / ngroups) << 4;
  const int col0 = (int)(t % ngroups) * (16 * NB);
  const int lm  = lane & 15;                 // M index (A) / N index (B)
  const int khi = lane >> 4;                 // which K half this lane holds
  const float* Arow = A + (long)(row0 + lm) * K + 2 * khi;
  const float* Bbase = B + col0 + lm;

  v8f acc[NB];
#pragma unroll
  for (int j = 0; j < NB; ++j) acc[j] = (v8f){};

  for (int k0 = 0; k0 < K; k0 += 4) {
    const v2f a = *(const v2f*)(Arow + k0);                 // 8B aligned
    const float* bp = Bbase + (long)(k0 + 2 * khi) * N;
#pragma unroll
    for (int j = 0; j < NB; ++j) {
      v2f b;
      b.x = bp[16 * j];
      b.y = bp[16 * j + N];
      acc[j] = __builtin_amdgcn_wmma_f32_16x16x4_f32(false, a, false, b,
                                                     (short)0, acc[j],
                                                     false, false);
    }
  }

  float* Crow = C + (long)(row0 + 8 * khi) * N + col0 + lm;
#pragma unroll
  for (int j = 0; j < NB; ++j)
#pragma unroll
    for (int r = 0; r < 8; ++r)
      Crow[(long)r * N + 16 * j] = acc[j][r];
}

// es[n,h] = dot(h[n,h,:], a_src[h,:]); ed likewise.
// One wave per (node,head); lanes stride channels (coalesced), shuffle-reduce.
__global__ void attn_logits(const float* __restrict__ h,
                            const float* __restrict__ asrc,
                            const float* __restrict__ adst,
                            float* __restrict__ es, float* __restrict__ ed,
                            int n, int heads, int ch) {
  const int lane = threadIdx.x & 31;
  long i = (long)blockIdx.x * (blockDim.x >> 5) + (threadIdx.x >> 5);
  if (i >= (long)n * heads) return;
  int hh = (int)(i % heads);
  long node = i / heads;
  const float* hp = h + node * (long)(heads * ch) + (long)hh * ch;
  const float* as = asrc + (long)hh * ch;
  const float* ad = adst + (long)hh * ch;
  float s = 0.f, d = 0.f;
  for (int c = lane; c < ch; c += 32) {
    float v = hp[c];
    s += v * as[c];
    d += v * ad[c];
  }
#pragma unroll
  for (int o = 16; o > 0; o >>= 1) {
    s += __shfl_down(s, o, 32);
    d += __shfl_down(d, o, 32);
  }
  if (lane == 0) { es[i] = s; ed[i] = d; }
}

__global__ void fill_f32(float* __restrict__ p, float v, long n) {
  long i = (long)blockIdx.x * blockDim.x + threadIdx.x;
  if (i < n) p[i] = v;
}
__global__ void fill_u32(unsigned* __restrict__ p, unsigned v, long n) {
  long i = (long)blockIdx.x * blockDim.x + threadIdx.x;
  if (i < n) p[i] = v;
}
// out[n][c] = b[c]  (bias folded into the accumulator init)
__global__ void init_bias(float* __restrict__ out, const float* __restrict__ b,
                          long n, int C) {
  long i = (long)blockIdx.x * blockDim.x + threadIdx.x;
  if (i < n * C) out[i] = b[i % C];
}
__global__ void elu_inplace(float* __restrict__ p, long n) {
  long i = (long)blockIdx.x * blockDim.x + threadIdx.x;
  if (i >= n) return;
  float v = p[i];
  p[i] = v > 0.f ? v : expm1f(v);
}

__device__ __forceinline__ void edge_sd(const int* __restrict__ ei, long e,
                                        int E, int& s, int& d) {
  if (e < E) { s = ei[e]; d = ei[(long)E + e]; }
  else       { s = d = (int)(e - E); }
}

// Pass A: segment max of leaky-relu logits (ordered-u32 key atomicMax).
__global__ void edge_max_k(const int* __restrict__ ei, int E, int n, int heads,
                           const float* __restrict__ es,
                           const float* __restrict__ ed,
                           unsigned* __restrict__ mkey) {
  long i = (long)blockIdx.x * blockDim.x + threadIdx.x;
  long tot = ((long)E + n) * heads;
  if (i >= tot) return;
  long e = i / heads; int hh = (int)(i % heads);
  int s, d; edge_sd(ei, e, E, s, d);
  float v = es[(long)s * heads + hh] + ed[(long)d * heads + hh];
  v = v > 0.f ? v : SLOPE * v;
  atomicMax(&mkey[(long)d * heads + hh], fkey(v));
}

// Pass B: denom[d,h] += exp(e - m[d,h])
__global__ void edge_den_k(const int* __restrict__ ei, int E, int n, int heads,
                           const float* __restrict__ es,
                           const float* __restrict__ ed,
                           const unsigned* __restrict__ mkey,
                           float* __restrict__ den) {
  long i = (long)blockIdx.x * blockDim.x + threadIdx.x;
  long tot = ((long)E + n) * heads;
  if (i >= tot) return;
  long e = i / heads; int hh = (int)(i % heads);
  int s, d; edge_sd(ei, e, E, s, d);
  float v = es[(long)s * heads + hh] + ed[(long)d * heads + hh];
  v = v > 0.f ? v : SLOPE * v;
  float m = unfkey(mkey[(long)d * heads + hh]);
  atomicAdd(&den[(long)d * heads + hh], __expf(v - m));
}

// Pass C: out[dst] += alpha * h[src]. One wave per edge; lanes stride channels
// (coalesced 128B accesses). heads <= 4.
__global__ void edge_aggr(const int* __restrict__ ei, int E, int n,
                          int heads, int ch,
                          const float* __restrict__ es,
                          const float* __restrict__ ed,
                          const unsigned* __restrict__ mkey,
                          const float* __restrict__ den,
                          const float* __restrict__ h,
                          float* __restrict__ out) {
  const int lane = threadIdx.x & 31;
  const long e = (long)blockIdx.x * (blockDim.x >> 5) + (threadIdx.x >> 5);
  if (e >= (long)E + n) return;
  int s, d; edge_sd(ei, e, E, s, d);
  float alp[4];
  for (int hh = 0; hh < heads; ++hh) {
    float v = es[(long)s * heads + hh] + ed[(long)d * heads + hh];
    v = v > 0.f ? v : SLOPE * v;
    float m = unfkey(mkey[(long)d * heads + hh]);
    float ex = __expf(v - m);
    alp[hh] = ex / (den[(long)d * heads + hh] + 1e-16f);
  }
  const int C = heads * ch;
  const float* hs = h + (long)s * C;
  float* od = out + (long)d * C;
  for (int c = lane; c < C; c += 32)
    atomicAdd(&od[c], alp[c / ch] * hs[c]);
}

extern "C" void kernel_launch(void* const* d_in, const int* in_sizes, int n_in,
                              void* d_out, int out_size, void* d_ws, size_t ws_size,
                              hipStream_t stream) {
  const float* x   = (const float*)d_in[0];
  const int*   ei  = (const int*)d_in[1];
  const float* W1  = (const float*)d_in[2];
  const float* as1 = (const float*)d_in[3];
  const float* ad1 = (const float*)d_in[4];
  const float* b1  = (const float*)d_in[5];
  const float* W2  = (const float*)d_in[6];
  const float* as2 = (const float*)d_in[7];
  const float* ad2 = (const float*)d_in[8];
  const float* b2  = (const float*)d_in[9];
  float* out = (float*)d_out;

  const int IN_DIM = 128, HEADS = 4, HID = 64, OUT_DIM = 32;
  const int C1 = HEADS * HID;            // 256
  const int N = in_sizes[0] / IN_DIM;    // 100000
  const int E = in_sizes[1] / 2;         // 1600000
  const long EN = (long)E + N;

  // workspace carve-up (fp32 units)
  float* ws = (float*)d_ws;
  float*    h1   = ws;             ws += (long)N * C1;
  float*    agg1 = ws;             ws += (long)N * C1;
  float*    es1  = ws;             ws += (long)N * HEADS;
  float*    ed1  = ws;             ws += (long)N * HEADS;
  unsigned* mk1  = (unsigned*)ws;  ws += (long)N * HEADS;
  float*    dn1  = ws;             ws += (long)N * HEADS;
  float*    h2   = ws;             ws += (long)N * OUT_DIM;
  float*    es2  = ws;             ws += N;
  float*    ed2  = ws;             ws += N;
  unsigned* mk2  = (unsigned*)ws;  ws += N;
  float*    dn2  = ws;             ws += N;

  const int BT = 256;
  const unsigned NEGINF_KEY = 0x007FFFFFu;   // fkey(-inf)
  auto g1 = [&](long n) { return dim3((unsigned)((n + BT - 1) / BT)); };
  auto gw = [&](long waves) { return dim3((unsigned)((waves + 7) / 8)); };

  // ---------------- Layer 1 ----------------
  long T1 = ((long)N >> 4) * (C1 / 64);      // 16x64 tiles per wave (NB=4)
  gemm_f32_wmma<4><<<gw(T1), BT, 0, stream>>>(x, W1, h1, N, IN_DIM, C1);
  attn_logits<<<gw((long)N * HEADS), BT, 0, stream>>>(h1, as1, ad1, es1, ed1,
                                                      N, HEADS, HID);
  fill_u32<<<g1((long)N * HEADS), BT, 0, stream>>>(mk1, NEGINF_KEY, (long)N * HEADS);
  fill_f32<<<g1((long)N * HEADS), BT, 0, stream>>>(dn1, 0.f, (long)N * HEADS);
  init_bias<<<g1((long)N * C1), BT, 0, stream>>>(agg1, b1, N, C1);
  edge_max_k<<<g1(EN * HEADS), BT, 0, stream>>>(ei, E, N, HEADS, es1, ed1, mk1);
  edge_den_k<<<g1(EN * HEADS), BT, 0, stream>>>(ei, E, N, HEADS, es1, ed1, mk1, dn1);
  edge_aggr<<<gw(EN), BT, 0, stream>>>(ei, E, N, HEADS, HID, es1, ed1, mk1, dn1,
                                       h1, agg1);
  elu_inplace<<<g1((long)N * C1), BT, 0, stream>>>(agg1, (long)N * C1);

  // ---------------- Layer 2 ----------------
  long T2 = ((long)N >> 4) * (OUT_DIM / 32); // 16x32 tiles per wave (NB=2)
  gemm_f32_wmma<2><<<gw(T2), BT, 0, stream>>>(agg1, W2, h2, N, C1, OUT_DIM);
  attn_logits<<<gw(N), BT, 0, stream>>>(h2, as2, ad2, es2, ed2, N, 1, OUT_DIM);
  fill_u32<<<g1(N), BT, 0, stream>>>(mk2, NEGINF_KEY, N);
  fill_f32<<<g1(N), BT, 0, stream>>>(dn2, 0.f, N);
  init_bias<<<g1((long)N * OUT_DIM), BT, 0, stream>>>(out, b2, N, OUT_DIM);
  edge_max_k<<<g1(EN), BT, 0, stream>>>(ei, E, N, 1, es2, ed2, mk2);
  edge_den_k<<<g1(EN), BT, 0, stream>>>(ei, E, N, 1, es2, ed2, mk2, dn2);
  edge_aggr<<<gw(EN), BT, 0, stream>>>(ei, E, N, 1, OUT_DIM, es2, ed2, mk2, dn2,
                                       h2, out);
}